// GNN_20151986553191
// MI455X (gfx1250) — compile-verified
//
#include <hip/hip_runtime.h>

typedef float v2f __attribute__((ext_vector_type(2)));
typedef float v8f __attribute__((ext_vector_type(8)));

static __device__ __forceinline__ int imin(int a, int b) { return a < b ? a : b; }

#define WMMA_F32_K4(a, b, c) \
  __builtin_amdgcn_wmma_f32_16x16x4_f32(false, (a), false, (b), (short)0, (c), false, false)

// ---------------------------------------------------------------- fill zeros (vectorized b128)
__global__ void gnn_zero_kernel(float* __restrict__ p, int n) {
  int i4 = (blockIdx.x * blockDim.x + threadIdx.x) * 4;
  if (i4 + 3 < n) {
    *(float4*)(p + i4) = make_float4(0.f, 0.f, 0.f, 0.f);
  } else {
    for (int j = i4; j < n; ++j) p[j] = 0.0f;
  }
}

// ---------------------------------------------------------------- degree of target nodes
__global__ void gnn_deg_kernel(const int* __restrict__ ei, float* __restrict__ deg, int nEdges) {
  int e = blockIdx.x * blockDim.x + threadIdx.x;
  if (e < nEdges) {
    int col = ei[nEdges + e];
    atomicAdd(&deg[col], 1.0f);
  }
}

// ---------------------------------------------------------------- layer1: dinv + s1 = (dinv*x)@W1   (WMMA f32 16x16x4, K padded 3->4)
__global__ __launch_bounds__(32) void gnn_layer1_wmma(const float* __restrict__ x,
                                                      const float* __restrict__ W1,
                                                      float* __restrict__ dinv,   // in: deg, out: rsqrt(deg+1)
                                                      float* __restrict__ s1,
                                                      int nNodes) {
  __shared__ float ldsDinv[16];
  const int lane = threadIdx.x;
  const int base = blockIdx.x * 16;
  const bool full = (base + 16 <= nNodes);   // uniform across the wave

  if (lane < 16) {
    int node = imin(base + lane, nNodes - 1);
    float di = rsqrtf(dinv[node] + 1.0f);    // self loop -> deg+1, always > 0
    ldsDinv[lane] = di;
    if (base + lane < nNodes) dinv[node] = di;
  }
  __syncthreads();

  const int h = lane >> 4;   // half-wave
  const int n = lane & 15;   // B/D column; also A row m
  const int m = n;
  const int am = imin(base + m, nNodes - 1);
  const int k0 = 2 * h;      // K slice start: 0 or 2
  const float di = ldsDinv[m];

  v2f a, b;
  a.x = x[am * 3 + k0] * di;                                // k0 in {0,2} always valid
  a.y = (k0 + 1 < 3) ? x[am * 3 + k0 + 1] * di : 0.0f;      // K=3 padded with zero
  b.x = W1[k0 * 16 + n];
  b.y = (k0 + 1 < 3) ? W1[(k0 + 1) * 16 + n] : 0.0f;

  v8f c = {};
  c = WMMA_F32_K4(a, b, c);

  if (full) {
    float* o = s1 + (size_t)(base + 8 * h) * 16 + n;
#pragma unroll
    for (int r = 0; r < 8; ++r) o[r * 16] = c[r];
  } else {
#pragma unroll
    for (int r = 0; r < 8; ++r) {
      int row = base + 8 * h + r;
      if (row < nNodes) s1[row * 16 + n] = c[r];
    }
  }
}

// ---------------------------------------------------------------- per-edge scatter: acc[col] += s[row]  (16 lanes per edge)
__global__ void gnn_edge_scatter(const int* __restrict__ ei,
                                 const float* __restrict__ s,
                                 float* __restrict__ acc,
                                 int nEdges) {
  int t = blockIdx.x * blockDim.x + threadIdx.x;
  int e = t >> 4;
  int f = t & 15;
  if (e < nEdges) {
    int row = ei[e];
    int col = ei[nEdges + e];
    atomicAdd(&acc[col * 16 + f], s[row * 16 + f]);
  }
}

// ---------------------------------------------------------------- layer2: h1 = relu(dinv*(acc1+s1)+b1);  s2 = (dinv*h1)@W2  (4x WMMA)
__global__ __launch_bounds__(32) void gnn_layer2_wmma(const float* __restrict__ s1,
                                                      const float* __restrict__ acc1,
                                                      const float* __restrict__ dinv,
                                                      const float* __restrict__ b1,
                                                      const float* __restrict__ W2,
                                                      float* __restrict__ s2,
                                                      int nNodes) {
  __shared__ float Hs[16][17];   // holds dinv-pre-scaled relu(h1); padded stride
  __shared__ float ldsDinv[16];
  const int lane = threadIdx.x;
  const int base = blockIdx.x * 16;
  const bool full = (base + 16 <= nNodes);   // uniform

  if (lane < 16) ldsDinv[lane] = dinv[imin(base + lane, nNodes - 1)];
  __syncthreads();

#pragma unroll
  for (int i = 0; i < 8; ++i) {
    int idx = lane + 32 * i;                 // 0..255 -> (local node, feature)
    int ln = idx >> 4, f = idx & 15;
    int node = full ? (base + ln) : imin(base + ln, nNodes - 1);
    float di = ldsDinv[ln];
    float v = di * (acc1[node * 16 + f] + s1[node * 16 + f]) + b1[f];
    Hs[ln][f] = fmaxf(v, 0.0f) * di;         // fold source-side dinv into A rows
  }
  __syncthreads();

  const int h = lane >> 4;
  const int n = lane & 15;
  const int m = n;

  v8f c = {};
#pragma unroll
  for (int k0 = 0; k0 < 16; k0 += 4) {
    v2f a, b;
    a.x = Hs[m][k0 + 2 * h + 0];
    a.y = Hs[m][k0 + 2 * h + 1];
    b.x = W2[(k0 + 2 * h + 0) * 16 + n];
    b.y = W2[(k0 + 2 * h + 1) * 16 + n];
    c = WMMA_F32_K4(a, b, c);
  }

  if (full) {
    float* o = s2 + (size_t)(base + 8 * h) * 16 + n;
#pragma unroll
    for (int r = 0; r < 8; ++r) o[r * 16] = c[r];
  } else {
#pragma unroll
    for (int r = 0; r < 8; ++r) {
      int row = base + 8 * h + r;
      if (row < nNodes) s2[row * 16 + n] = c[r];
    }
  }
}

// ---------------------------------------------------------------- finalize layer2 + global_add_pool
__global__ void gnn_pool_kernel(const float* __restrict__ s2,
                                const float* __restrict__ acc2,
                                const float* __restrict__ dinv,
                                const float* __restrict__ b2,
                                const int* __restrict__ batch,
                                float* __restrict__ pooled,
                                int nNodes) {
  int t = blockIdx.x * blockDim.x + threadIdx.x;
  int node = t >> 4;
  int f = t & 15;
  if (node < nNodes) {
    float v = dinv[node] * (acc2[node * 16 + f] + s2[node * 16 + f]) + b2[f];
    atomicAdd(&pooled[batch[node] * 16 + f], v);
  }
}

// ---------------------------------------------------------------- logits = pooled@Wl + bl (WMMA, N padded 7->16) then log_softmax
__global__ __launch_bounds__(32) void gnn_logits_wmma(const float* __restrict__ pooled,
                                                      const float* __restrict__ Wl,
                                                      const float* __restrict__ bl,
                                                      float* __restrict__ out,
                                                      int nGraphs) {
  __shared__ float L[16][17];
  const int lane = threadIdx.x;
  const int base = blockIdx.x * 16;
  const int h = lane >> 4;
  const int n = lane & 15;
  const int m = n;
  const int gm = imin(base + m, nGraphs - 1);

  v8f c = {};
#pragma unroll
  for (int k0 = 0; k0 < 16; k0 += 4) {
    v2f a = *(const v2f*)&pooled[gm * 16 + k0 + 2 * h];   // even offset -> 8B aligned
    v2f b;
    b.x = (n < 7) ? Wl[(k0 + 2 * h + 0) * 7 + n] : 0.0f;
    b.y = (n < 7) ? Wl[(k0 + 2 * h + 1) * 7 + n] : 0.0f;
    c = WMMA_F32_K4(a, b, c);
  }

  float bias = (n < 7) ? bl[n] : 0.0f;
#pragma unroll
  for (int r = 0; r < 8; ++r) L[8 * h + r][n] = c[r] + bias;
  __syncthreads();

  if (lane < 16) {
    int g = base + lane;
    if (g < nGraphs) {
      float mx = L[lane][0];
#pragma unroll
      for (int j = 1; j < 7; ++j) mx = fmaxf(mx, L[lane][j]);
      float s = 0.0f;
#pragma unroll
      for (int j = 0; j < 7; ++j) s += expf(L[lane][j] - mx);
      float lse = mx + logf(s);
#pragma unroll
      for (int j = 0; j < 7; ++j) out[g * 7 + j] = L[lane][j] - lse;
    }
  }
}

// ================================================================ launcher
extern "C" void kernel_launch(void* const* d_in, const int* in_sizes, int n_in,
                              void* d_out, int out_size, void* d_ws, size_t ws_size,
                              hipStream_t stream) {
  const float* x  = (const float*)d_in[0];
  const int*   ei = (const int*)d_in[1];
  const int*   batch = (const int*)d_in[2];
  const float* W1 = (const float*)d_in[3];
  const float* b1 = (const float*)d_in[4];
  const float* W2 = (const float*)d_in[5];
  const float* b2 = (const float*)d_in[6];
  const float* Wl = (const float*)d_in[7];
  const float* bl = (const float*)d_in[8];
  float* out = (float*)d_out;

  const int N = in_sizes[0] / 3;   // 100000
  const int E = in_sizes[1] / 2;   // 3200000
  const int G = out_size / 7;      // 1024

  // workspace layout (floats): [deg/dinv N][acc1 16N][acc2 16N][pooled 16G][s1 16N][s2 16N]
  float* ws     = (float*)d_ws;
  float* dinv   = ws;
  float* acc1   = dinv + N;
  float* acc2   = acc1 + (size_t)16 * N;
  float* pooled = acc2 + (size_t)16 * N;
  float* s1     = pooled + (size_t)16 * G;
  float* s2     = s1 + (size_t)16 * N;

  const int zn = N + 32 * N + 16 * G;   // deg + acc1 + acc2 + pooled (contiguous)
  gnn_zero_kernel<<<(zn / 4 + 255) / 256, 256, 0, stream>>>(ws, zn);

  gnn_deg_kernel<<<(E + 255) / 256, 256, 0, stream>>>(ei, dinv, E);

  const int nodeTiles = (N + 15) / 16;
  gnn_layer1_wmma<<<nodeTiles, 32, 0, stream>>>(x, W1, dinv, s1, N);

  const int edgeThreads = E * 16;
  gnn_edge_scatter<<<(edgeThreads + 255) / 256, 256, 0, stream>>>(ei, s1, acc1, E);

  gnn_layer2_wmma<<<nodeTiles, 32, 0, stream>>>(s1, acc1, dinv, b1, W2, s2, N);

  gnn_edge_scatter<<<(edgeThreads + 255) / 256, 256, 0, stream>>>(ei, s2, acc2, E);

  gnn_pool_kernel<<<(N * 16 + 255) / 256, 256, 0, stream>>>(s2, acc2, dinv, b2, batch, pooled, N);

  gnn_logits_wmma<<<(G + 15) / 16, 32, 0, stream>>>(pooled, Wl, bl, out, G);
}